// ScaledDotProduct_13975823581382
// MI455X (gfx1250) — compile-verified
//
#include <hip/hip_runtime.h>

typedef __attribute__((ext_vector_type(2))) float v2f;
typedef __attribute__((ext_vector_type(4))) float v4f;
typedef __attribute__((ext_vector_type(8))) float v8f;

constexpr int kH = 16;
constexpr int kS = 4096;
constexpr int kD = 64;
constexpr int kWaves = 4;
constexpr int kBlockQ = kWaves * 16;   // 64 query rows per workgroup
constexpr float kScale = 0.125f;       // 1 / TEMPERATURE (= 1/8)

// Low 32 bits of a generic pointer to LDS == LDS byte address (flat LDS aperture).
static __device__ __forceinline__ uint32_t lds_addr32(const void* p) {
  return (uint32_t)(size_t)p;
}

// Stage a 16x64 f32 tile (4 KB, contiguous) global -> LDS via CDNA5 async path.
// 128 threads x 2 transfers x 16B = 4 KB. Tracked by ASYNCcnt.
static __device__ __forceinline__ void async_stage_tile(uint32_t lds_base,
                                                        const float* gbase,
                                                        int tid) {
  asm volatile(
      "global_load_async_to_lds_b128 %0, %2, off\n\t"
      "global_load_async_to_lds_b128 %1, %3, off"
      :
      : "v"(lds_base + (uint32_t)tid * 16u),
        "v"(lds_base + (uint32_t)(tid + 128) * 16u),
        "v"(gbase + (size_t)tid * 4),
        "v"(gbase + (size_t)(tid + 128) * 4)
      : "memory");
}

static __device__ __forceinline__ void wait_async0() {
  asm volatile("s_wait_asynccnt 0x0" ::: "memory");
}

__global__ __launch_bounds__(kWaves * 32)
void fa_causal_f32wmma(const float* __restrict__ q,
                       const float* __restrict__ k,
                       const float* __restrict__ v,
                       float* __restrict__ out,
                       float* __restrict__ attn) {
  __shared__ float sK[2][16 * kD];            // double-buffered key tiles (8 KB)
  __shared__ float sV[2][16 * kD];            // double-buffered value tiles (8 KB)
  __shared__ float sP[kWaves][16][16];        // per-wave P tiles (4 KB)

  const int wg    = blockIdx.x;
  const int h     = wg / (kS / kBlockQ);
  const int q0wg  = (wg % (kS / kBlockQ)) * kBlockQ;
  const int tid   = threadIdx.x;
  const int wave  = tid >> 5;
  const int lane  = tid & 31;
  const int hi    = lane >> 4;     // half-wave selector
  const int ln    = lane & 15;
  const int q0    = q0wg + wave * 16;          // this wave's first query row
  const int jmaxw = q0 + 15;                   // last key this wave needs

  const uint32_t sK_lds[2] = {lds_addr32(&sK[0][0]), lds_addr32(&sK[1][0])};
  const uint32_t sV_lds[2] = {lds_addr32(&sV[0][0]), lds_addr32(&sV[1][0])};
  const float* kh = k + (size_t)h * kS * kD;
  const float* vh = v + (size_t)h * kS * kD;

  // ---- Q fragments: A-matrix layout for V_WMMA_F32_16X16X4_F32 ----
  // lanes 0-15 hold M=lane rows with K = 4kk,4kk+1 ; lanes 16-31 K = 4kk+2,4kk+3
  v2f aq[16];
  {
    const float* qrow = q + ((size_t)h * kS + q0 + ln) * kD;
#pragma unroll
    for (int kk = 0; kk < 16; ++kk) {
      const int dbase = kk * 4 + hi * 2;
      aq[kk] = *(const v2f*)(qrow + dbase);
    }
  }

  // per-lane row stats: lanes 0-15 own rows q0+0..7, lanes 16-31 rows q0+8..15
  float m[8], l[8];
#pragma unroll
  for (int r = 0; r < 8; ++r) { m[r] = -__builtin_inff(); l[r] = 0.f; }

  const int nblkWG = (q0wg + kBlockQ) / 16;    // causal key-blocks for workgroup

  // =================== Pass 1: online softmax stats ===================
  async_stage_tile(sK_lds[0], kh, tid);        // prefetch K tile 0
  wait_async0();
  __syncthreads();

  for (int jb = 0; jb < nblkWG; ++jb) {
    const int j0  = jb * 16;
    const int cur = jb & 1;
    if (jb + 1 < nblkWG)                       // prefetch next K tile (other buf)
      async_stage_tile(sK_lds[1 - cur], kh + (size_t)(j0 + 16) * kD, tid);

    if (j0 <= jmaxw) {                         // wave-uniform: EXEC all-ones
      v8f c = {};
#pragma unroll
      for (int kk = 0; kk < 16; ++kk) {
        const int dbase = kk * 4 + hi * 2;
        v2f b = *(const v2f*)(&sK[cur][ln * kD + dbase]);  // B(d,n)=K[j0+n][d]
        c = __builtin_amdgcn_wmma_f32_16x16x4_f32(false, aq[kk], false, b,
                                                  (short)0, c, false, false);
      }
#pragma unroll
      for (int r = 0; r < 8; ++r) {
        const int row = q0 + r + 8 * hi;
        float sv = (j0 + ln > row) ? -__builtin_inff() : c[r] * kScale;
        float tm = sv;                                    // 16-lane row max
        tm = fmaxf(tm, __shfl_xor(tm, 1, 32));
        tm = fmaxf(tm, __shfl_xor(tm, 2, 32));
        tm = fmaxf(tm, __shfl_xor(tm, 4, 32));
        tm = fmaxf(tm, __shfl_xor(tm, 8, 32));
        const float mn = fmaxf(m[r], tm);
        float p = __expf(sv - mn);                        // exp(-inf)=0 on mask
        float rs = p;                                     // 16-lane row sum
        rs += __shfl_xor(rs, 1, 32);
        rs += __shfl_xor(rs, 2, 32);
        rs += __shfl_xor(rs, 4, 32);
        rs += __shfl_xor(rs, 8, 32);
        l[r] = l[r] * __expf(m[r] - mn) + rs;
        m[r] = mn;
      }
    }
    wait_async0();                             // next buffer resident
    __syncthreads();                           // all waves done with cur buffer
  }

  float invl[8];
#pragma unroll
  for (int r = 0; r < 8; ++r) invl[r] = 1.f / l[r];

  // =================== Pass 2: P store + O = P·V ===================
  v8f o[4] = {};                               // D split into 4x16 columns
  async_stage_tile(sK_lds[0], kh, tid);        // prefetch K/V tile 0
  async_stage_tile(sV_lds[0], vh, tid);
  wait_async0();
  __syncthreads();

  for (int jb = 0; jb < nblkWG; ++jb) {
    const int j0  = jb * 16;
    const int cur = jb & 1;
    if (jb + 1 < nblkWG) {                     // prefetch next K/V tiles
      async_stage_tile(sK_lds[1 - cur], kh + (size_t)(j0 + 16) * kD, tid);
      async_stage_tile(sV_lds[1 - cur], vh + (size_t)(j0 + 16) * kD, tid);
    }

    if (j0 <= jmaxw) {
      v8f c = {};
#pragma unroll
      for (int kk = 0; kk < 16; ++kk) {
        const int dbase = kk * 4 + hi * 2;
        v2f b = *(const v2f*)(&sK[cur][ln * kD + dbase]);
        c = __builtin_amdgcn_wmma_f32_16x16x4_f32(false, aq[kk], false, b,
                                                  (short)0, c, false, false);
      }
      // normalized probabilities: stream to attn (nontemporal) + LDS for PV
#pragma unroll
      for (int r = 0; r < 8; ++r) {
        const int row = q0 + r + 8 * hi;
        const float sv = (j0 + ln > row) ? -__builtin_inff() : c[r] * kScale;
        const float p  = __expf(sv - m[r]) * invl[r];
        __builtin_nontemporal_store(p, attn + ((size_t)h * kS + row) * kS + j0 + ln);
        sP[wave][r + 8 * hi][ln] = p;
      }
      asm volatile("s_wait_dscnt 0x0" ::: "memory");  // cross-lane LDS RAW fence

      // O += P(16x16) * V(16x64): 4 K-chunks x 4 D-chunks of f32 WMMA
#pragma unroll
      for (int t = 0; t < 4; ++t) {
        v2f a;
        a.x = sP[wave][ln][t * 4 + 2 * hi];
        a.y = sP[wave][ln][t * 4 + 2 * hi + 1];
#pragma unroll
        for (int dc = 0; dc < 4; ++dc) {
          v2f b;
          b.x = sV[cur][(t * 4 + 2 * hi)     * kD + dc * 16 + ln];
          b.y = sV[cur][(t * 4 + 2 * hi + 1) * kD + dc * 16 + ln];
          o[dc] = __builtin_amdgcn_wmma_f32_16x16x4_f32(false, a, false, b,
                                                        (short)0, o[dc], false, false);
        }
      }
    }
    wait_async0();
    __syncthreads();
  }

  // ---- zero-fill strictly-upper attn region (streaming float4 stores) ----
  {
    const int zstart = q0 + 16;                // 16-aligned; diag block wrote 0s
    const int n4 = (kS - zstart) >> 2;
    const v4f z = {};
    for (int r = 0; r < 16; ++r) {
      v4f* zp = (v4f*)(attn + ((size_t)h * kS + q0 + r) * kS + zstart);
      for (int idx = lane; idx < n4; idx += 32)
        __builtin_nontemporal_store(z, zp + idx);
    }
  }

  // ---- write output tile ----
#pragma unroll
  for (int dc = 0; dc < 4; ++dc) {
#pragma unroll
    for (int r = 0; r < 8; ++r) {
      const int row = q0 + r + 8 * hi;
      out[((size_t)h * kS + row) * kD + dc * 16 + ln] = o[dc][r];
    }
  }
}

extern "C" void kernel_launch(void* const* d_in, const int* in_sizes, int n_in,
                              void* d_out, int out_size, void* d_ws, size_t ws_size,
                              hipStream_t stream) {
  const float* q = (const float*)d_in[0];
  const float* k = (const float*)d_in[1];
  const float* v = (const float*)d_in[2];
  float* out  = (float*)d_out;
  float* attn = out + (size_t)kH * kS * kD;   // outputs concatenated: [O | attn]

  dim3 grid(kH * (kS / kBlockQ));             // 16 heads * 64 query blocks = 1024
  dim3 block(kWaves * 32);                    // 4 waves (wave32)
  hipLaunchKernelGGL(fa_causal_f32wmma, grid, block, 0, stream, q, k, v, out, attn);
}